// SoftQuantizer_52286931862176
// MI455X (gfx1250) — compile-verified
//
#include <hip/hip_runtime.h>
#include <hip/hip_bf16.h>

// Soft quantizer (STE): per element of x (6.29M fp32), softmax over negative
// squared distances to 25 uniform levels in [-2,2]; outputs (x_ste, x_hard,
// symbols) concatenated in d_out (3 * N four-byte slots).
//
// MI455X reasoning: ~100 MB of traffic -> ~4.3 us floor @ 23.3 TB/s.
// Inner loop pre-scaled by sqrt(log2 e): 4 VALU + 1 v_exp_f32 per level
// (TRANS co-executes with VALU), 8 elements/thread for ILP + amortization.
// WMMA is not applicable (elementwise softmax; lane-striped D layout would
// force cross-lane reductions costlier than the FMAs saved). CDNA5-specific
// data path: async global->LDS staging via global_load_async_to_lds_b128
// (imm-offset form) + s_wait_asynccnt, LDS readback via ds_load_b128.

#define LVL 25
#define MINV (-2.0f)
#define INV_STEP 6.0f            // (LVL-1) / (MAX-MIN) = 24/4
#define LOG2E 1.44269504088896340736f
#define SQRT_LOG2E 1.20112240878644980f   // sqrt(log2 e)
#define EPT 8                    // elements per thread

#if defined(__gfx1250__) && __has_builtin(__builtin_amdgcn_global_load_async_to_lds_b128) && __has_builtin(__builtin_amdgcn_s_wait_asynccnt)
#define USE_ASYNC_LDS 1
#else
#define USE_ASYNC_LDS 0
#endif

#if USE_ASYNC_LDS
// Builtin signature (probe-verified by clang diagnostic): param0 = int4-vector
// pointer in global (addrspace(1)) AS; param1 = LDS side; imm offset; imm cpol.
typedef int v4i_async __attribute__((vector_size(16)));
typedef __attribute__((address_space(1))) v4i_async* g_v4i_p;
typedef __attribute__((address_space(3))) v4i_async* l_v4i_p;
#endif

__device__ __forceinline__ float fast_exp2(float t) {
#if defined(__gfx1250__) && __has_builtin(__builtin_amdgcn_exp2f)
  return __builtin_amdgcn_exp2f(t);
#else
  return exp2f(t);
#endif
}

__device__ __forceinline__ float fast_rcp(float t) {
#if defined(__gfx1250__) && __has_builtin(__builtin_amdgcn_rcpf)
  return __builtin_amdgcn_rcpf(t);
#else
  return 1.0f / t;
#endif
}

// One element: analytic nearest level (uniform grid) refined over {i-1,i,i+1}
// with the exact levels[] values so symbols/x_hard match jnp.argmin
// (first-index tie rule via strict '<' on ascending candidate indices).
__device__ __forceinline__ void quantize1(float xe,
                                          const float* __restrict__ levels,
                                          const float lev[LVL],
                                          const float levk[LVL],
                                          float& ste, float& hard, int& sym) {
  float fpos = (xe - MINV) * INV_STEP;
  int i1 = (int)__builtin_rintf(fpos);
  i1 = i1 < 0 ? 0 : (i1 > LVL - 1 ? LVL - 1 : i1);
  int i0 = i1 > 0 ? i1 - 1 : 0;
  int i2 = i1 < LVL - 1 ? i1 + 1 : LVL - 1;
  // L0-resident per-lane gathers of the exact reference level values.
  float c0 = levels[i0], c1 = levels[i1], c2 = levels[i2];
  float e0 = xe - c0, e1 = xe - c1, e2 = xe - c2;
  float d0 = e0 * e0, d1 = e1 * e1, d2 = e2 * e2;
  int idx = i0; float dm = d0; float ch = c0;
  if (d1 < dm) { idx = i1; dm = d1; ch = c1; }   // strict: first index wins ties
  if (d2 < dm) { idx = i2; dm = d2; ch = c2; }

  // softmax(-(d_l)) with max subtraction == exp(dm - d_l); SIGMA == 1.
  // Pre-scaled into exp2 domain: exp2(dm*log2e - ((x - c_l)*sqrt(log2e))^2).
  float xk = xe * SQRT_LOG2E;
  float dmk = dm * LOG2E;
  float sum = 0.0f, wsum = 0.0f;
#pragma unroll
  for (int l = 0; l < LVL; ++l) {
    float dfk = xk - levk[l];
    float t = __builtin_fmaf(-dfk, dfk, dmk);    // dmk - dfk*dfk  (<= ~0)
    float el = fast_exp2(t);                     // v_exp_f32
    sum += el;
    wsum = __builtin_fmaf(lev[l], el, wsum);
  }
  float xsoft = wsum * fast_rcp(sum);
  hard = ch;
  sym = idx;
  ste = xsoft + (ch - xsoft);                    // matches reference STE forward
}

__global__ __launch_bounds__(256) void softq_kernel(
    const float* __restrict__ x, const float* __restrict__ levels,
    float* __restrict__ out_ste, float* __restrict__ out_hard,
    int* __restrict__ out_sym, int n) {
#if USE_ASYNC_LDS
  __shared__ float4 tile[256 * (EPT / 4)];
#endif
  // Hoist levels into SGPRs (uniform address, constant indices -> s_load),
  // plus the sqrt(log2e)-scaled copy used by the exp2-domain inner loop.
  float lev[LVL], levk[LVL];
#pragma unroll
  for (int l = 0; l < LVL; ++l) {
    lev[l] = levels[l];
    levk[l] = lev[l] * SQRT_LOG2E;
  }

  const int tid = blockIdx.x * 256 + threadIdx.x;
  const int base = tid * EPT;
  if (base >= n) return;

  if (base + EPT - 1 < n) {
    float4 xv[EPT / 4];
#if USE_ASYNC_LDS
    {
      g_v4i_p gp = (g_v4i_p)(const_cast<float*>(x) + base);
      l_v4i_p lp = (l_v4i_p)(&tile[(EPT / 4) * threadIdx.x]);
      __builtin_amdgcn_global_load_async_to_lds_b128(gp, lp, 0, 0);
      __builtin_amdgcn_global_load_async_to_lds_b128(gp, lp, 16, 0);
      __builtin_amdgcn_s_wait_asynccnt(0);
      xv[0] = tile[(EPT / 4) * threadIdx.x + 0];
      xv[1] = tile[(EPT / 4) * threadIdx.x + 1];
    }
#else
    xv[0] = *(const float4*)(x + base);
    xv[1] = *(const float4*)(x + base + 4);
#endif
    float xs[EPT] = {xv[0].x, xv[0].y, xv[0].z, xv[0].w,
                     xv[1].x, xv[1].y, xv[1].z, xv[1].w};
    float ste[EPT], hard[EPT];
    int sym[EPT];
#pragma unroll
    for (int e = 0; e < EPT; ++e)
      quantize1(xs[e], levels, lev, levk, ste[e], hard[e], sym[e]);

#pragma unroll
    for (int v = 0; v < EPT / 4; ++v) {
      const int o = base + 4 * v;
      *(float4*)(out_ste + o) =
          make_float4(ste[4 * v], ste[4 * v + 1], ste[4 * v + 2], ste[4 * v + 3]);
      *(float4*)(out_hard + o) =
          make_float4(hard[4 * v], hard[4 * v + 1], hard[4 * v + 2], hard[4 * v + 3]);
      *(int4*)(out_sym + o) =
          make_int4(sym[4 * v], sym[4 * v + 1], sym[4 * v + 2], sym[4 * v + 3]);
    }
  } else {
    for (int j = base; j < n; ++j) {
      float ste, hard;
      int sym;
      quantize1(x[j], levels, lev, levk, ste, hard, sym);
      out_ste[j] = ste;
      out_hard[j] = hard;
      out_sym[j] = sym;
    }
  }
}

extern "C" void kernel_launch(void* const* d_in, const int* in_sizes, int n_in,
                              void* d_out, int out_size, void* d_ws, size_t ws_size,
                              hipStream_t stream) {
  const float* x = (const float*)d_in[0];
  const float* levels = (const float*)d_in[1];
  const int n = in_sizes[0];                 // 6,291,456
  float* out = (float*)d_out;                // [ste(n) | hard(n) | sym(n)]
  float* out_ste = out;
  float* out_hard = out + (size_t)n;
  int* out_sym = (int*)(out + 2 * (size_t)n);

  const int threads = 256;
  const int blocks = (n + threads * EPT - 1) / (threads * EPT);
  softq_kernel<<<blocks, threads, 0, stream>>>(x, levels, out_ste, out_hard,
                                               out_sym, n);
}